// EmbeddingToAffinities_48215302864984
// MI455X (gfx1250) — compile-verified
//
#include <hip/hip_runtime.h>
#include <stdint.h>

// Tensor geometry (B=1, C=16, D=32, H=256, W=256)
#define CSTRIDE 2097152u  // channel stride in elements (D*H*W)
#define DSTRIDE 65536u    // d stride (H*W)
#define HSTRIDE 256u
#define NOFF 15

// LDS norm-tile layout (odd strides -> conflict-free banks)
#define R0_ROWS 52
#define R0_STRIDE 97
#define R0_SIZE (R0_ROWS * R0_STRIDE)   // 5044
#define R1_ROWS 17
#define R1_STRIDE 69
#define R1_SIZE (R1_ROWS * R1_STRIDE)   // 1173
#define LDS_FLOATS (R0_SIZE + R1_SIZE)  // 6217 (~24.9 KB)

typedef float v2f __attribute__((ext_vector_type(2)));
typedef float v4f __attribute__((ext_vector_type(4)));

struct v4 { float x, y, z, w; };
struct __attribute__((packed, aligned(4))) f4u { float x, y, z, w; };

__device__ __forceinline__ int imin(int a, int b) { return a < b ? a : b; }
__device__ __forceinline__ int imax(int a, int b) { return a > b ? a : b; }

__device__ __forceinline__ void setc(v4& v, int j, float f) {
  if (j == 0) v.x = f; else if (j == 1) v.y = f; else if (j == 2) v.z = f; else v.w = f;
}

// Packed f32 FMA (V_PK_FMA_F32, VOP3P) -- explicit so the 2x f32/instr path is guaranteed.
__device__ __forceinline__ void pk_fma_acc(v2f& acc, v2f a, v2f b) {
  asm("v_pk_fma_f32 %0, %1, %2, %0" : "+v"(acc) : "v"(a), "v"(b));
}
__device__ __forceinline__ v2f pk_fma(v2f a, v2f b, v2f c) {
  v2f d;
  asm("v_pk_fma_f32 %0, %1, %2, %3" : "=v"(d) : "v"(a), "v"(b), "v"(c));
  return d;
}

// aff = 2/(1+exp(2*arg)) = 1 - tanh(arg); clamp floor == reference's e>=1e-6 clamp.
__device__ __forceinline__ float aff_from_arg(float arg) {
#if __has_builtin(__builtin_amdgcn_tanhf)
  return fmaxf(1.0f - __builtin_amdgcn_tanhf(arg), 1.999998000002e-6f);
#else
  float e = __builtin_amdgcn_exp2f(arg * -2.8853900817779268f);  // exp(-2*arg)
  e = fmaxf(e, 1e-6f);
  return (2.0f * e) * __builtin_amdgcn_rcpf(e + 1.0f);
#endif
}

// CDNA5 async global->LDS copy (ASYNCcnt path), portable inline asm.
__device__ __forceinline__ void async_g2l_b32(float* ldst, const float* gsrc) {
  uint32_t l = (uint32_t)(uintptr_t)ldst;  // low 32 bits = LDS byte offset
  uint64_t g = (uint64_t)(uintptr_t)gsrc;
  asm volatile("global_load_async_to_lds_b32 %0, %1, off" :: "v"(l), "v"(g) : "memory");
}
__device__ __forceinline__ void wait_async0() {
  asm volatile("s_wait_asynccnt 0" ::: "memory");
}

// NT store of a 4-voxel result (16B-aligned; keeps streamed output out of L2's
// way so the 134MB emb stays resident for the 15x neighbor gather).
__device__ __forceinline__ void store_nt4(float* p, const v4& a) {
  const v4f o = {a.x, a.y, a.z, a.w};
  __builtin_nontemporal_store(o, reinterpret_cast<v4f*>(p));
}

// Stage the norm tile + halos for planes d and d-1 into LDS.
template <bool CLAMP>
__device__ __forceinline__ void stage_norms(const float* __restrict__ Nrm, float* lds,
                                            int d, int h0, int w0, int tid) {
  const float* Np = Nrm + (size_t)d * DSTRIDE;
  for (int i = tid; i < R0_SIZE; i += 256) {
    const int r = i / R0_STRIDE, c = i - r * R0_STRIDE;
    int gh = h0 - 27 + r, gw = w0 - 27 + c;
    if (CLAMP) { gh = imin(imax(gh, 0), 255); gw = imin(imax(gw, 0), 255); }
    async_g2l_b32(&lds[i], Np + gh * HSTRIDE + gw);
  }
  const float* Npp = Nrm + (size_t)(CLAMP ? imax(d - 1, 0) : (d - 1)) * DSTRIDE;
  for (int i = tid; i < R1_SIZE; i += 256) {
    const int r = i / R1_STRIDE, c = i - r * R1_STRIDE;
    int gh = h0 - 1 + r, gw = w0 - 1 + c;
    if (CLAMP) { gh = imin(imax(gh, 0), 255); gw = imin(imax(gw, 0), 255); }
    async_g2l_b32(&lds[R0_SIZE + i], Npp + gh * HSTRIDE + gw);
  }
  wait_async0();
  __syncthreads();
}

// 16-channel packed dot (USE_N) or squared-diff (!USE_N) for a 4-voxel group.
// cp[c] = emb + sbase + c*CSTRIDE (hoisted); delta is a compile-time constant per
// offset after unrolling, so every load is  global_load_b128 v, v[cp_c], off offset:imm.
template <bool USE_N>
__device__ __forceinline__ void dot16p(const float* const* cp, int delta,
                                       const v2f* a01, const v2f* a23,
                                       v2f& acc01, v2f& acc23) {
  acc01 = v2f{0.0f, 0.0f};
  acc23 = v2f{0.0f, 0.0f};
#pragma unroll
  for (int c = 0; c < 16; ++c) {
    const f4u b = *reinterpret_cast<const f4u*>(cp[c] + delta);
    const v2f b01 = {b.x, b.y}, b23 = {b.z, b.w};
    if constexpr (USE_N) {
      pk_fma_acc(acc01, a01[c], b01);
      pk_fma_acc(acc23, a23[c], b23);
    } else {
      const v2f t01 = a01[c] - b01, t23 = a23[c] - b23;
      pk_fma_acc(acc01, t01, t01);
      pk_fma_acc(acc23, t23, t23);
    }
  }
}

__device__ __forceinline__ void aff_fast_useN(const float* lds, int bi,
                                              v2f halfN01, v2f halfN23,
                                              v2f acc01, v2f acc23, float coef, v4& aff) {
  const v2f half2 = {0.5f, 0.5f}, cf = {coef, coef};
  const v2f ny01 = {lds[bi + 0], lds[bi + 1]};
  const v2f ny23 = {lds[bi + 2], lds[bi + 3]};
  const v2f s01 = pk_fma(half2, ny01, halfN01);
  const v2f s23 = pk_fma(half2, ny23, halfN23);
  const v2f g01 = (s01 - acc01) * cf;
  const v2f g23 = (s23 - acc23) * cf;
  aff.x = aff_from_arg(g01.x); aff.y = aff_from_arg(g01.y);
  aff.z = aff_from_arg(g23.x); aff.w = aff_from_arg(g23.y);
}

// Offset tables
__device__ constexpr int   c_DD [NOFF] = {-1, 0, 0,  0, 0,  0, 0,  0,  0, 0, 0, -1, -1,   0, 0};
__device__ constexpr int   c_DH [NOFF] = { 0,-1, 0, -9, 0, -9, 9, -9, -4, 4, 9, -1,  0, -27, 0};
__device__ constexpr int   c_DW [NOFF] = { 0, 0,-1,  0,-9, -9,-9,  4, -9,-9,-4,  0, -1,   0,-27};
__device__ constexpr float c_S2 [NOFF] = { 1, 1, 1, 81,81,162,162, 97, 97,97,97,  2,  2, 729,729};

// ---------------- Kernel 1: per-voxel squared channel norms --------------
__global__ __launch_bounds__(256) void norm_kernel(const float* __restrict__ emb,
                                                   float* __restrict__ Nrm) {
  const uint32_t base = (blockIdx.x * 256u + threadIdx.x) * 4u;
  v2f a01 = {0.f, 0.f}, a23 = {0.f, 0.f};
#pragma unroll
  for (int c = 0; c < 16; ++c) {
    const f4u v = *reinterpret_cast<const f4u*>(emb + base + (uint32_t)c * CSTRIDE);
    const v2f v01 = {v.x, v.y}, v23 = {v.z, v.w};
    pk_fma_acc(a01, v01, v01);
    pk_fma_acc(a23, v23, v23);
  }
  const v4f o = {a01.x, a01.y, a23.x, a23.y};
  *reinterpret_cast<v4f*>(Nrm + base) = o;
}

// ---------------- Kernel 2a: interior blocks (no bounds checks at all) ---
__global__ __launch_bounds__(256) void aff_interior(const float* __restrict__ emb,
                                                    const float* __restrict__ Nrm,
                                                    float* __restrict__ out) {
  extern __shared__ float lds[];
  const int tid = threadIdx.x;
  const int w0 = 64 + blockIdx.x * 64;   // {64, 128}
  const int h0 = 32 + blockIdx.y * 16;   // 32..224
  const int d  = 1 + blockIdx.z;         // 1..31

  stage_norms<false>(Nrm, lds, d, h0, w0, tid);

  const int h = h0 + (tid >> 4);
  const int w = w0 + (tid & 15) * 4;
  const uint32_t sbase = (uint32_t)d * DSTRIDE + (uint32_t)h * HSTRIDE + (uint32_t)w;

  // Hoisted per-channel bases: all neighbor loads become imm-offset b128 loads.
  const float* cp[16];
  v2f a01[16], a23[16];
#pragma unroll
  for (int c = 0; c < 16; ++c) {
    cp[c] = emb + sbase + (uint32_t)c * CSTRIDE;
    const f4u t = *reinterpret_cast<const f4u*>(cp[c]);
    a01[c] = v2f{t.x, t.y};
    a23[c] = v2f{t.z, t.w};
  }

  const int sb = ((tid >> 4) + 27) * R0_STRIDE + ((tid & 15) * 4 + 27);
  const v2f halfN01 = {0.5f * lds[sb + 0], 0.5f * lds[sb + 1]};
  const v2f halfN23 = {0.5f * lds[sb + 2], 0.5f * lds[sb + 3]};

#pragma unroll
  for (int k = 0; k < NOFF; ++k) {
    const int dd = c_DD[k], dh = c_DH[k], dwo = c_DW[k];
    const float coef = 0.5f / c_S2[k];
    const int delta = dd * (int)DSTRIDE + dh * (int)HSTRIDE + dwo;  // constexpr after unroll
    v2f acc01, acc23;
    dot16p<true>(cp, delta, a01, a23, acc01, acc23);
    const int hn = (tid >> 4) + dh, wn = (tid & 15) * 4 + dwo;  // tile-relative
    const int bi = (dd == 0) ? ((hn + 27) * R0_STRIDE + (wn + 27))
                             : (R0_SIZE + (hn + 1) * R1_STRIDE + (wn + 1));
    v4 aff;
    aff_fast_useN(lds, bi, halfN01, halfN23, acc01, acc23, coef, aff);
    store_nt4(out + (uint32_t)k * CSTRIDE + sbase, aff);
  }
}

// ---------------- Kernel 2b: general (boundary-capable) ------------------
template <bool USE_N>
__global__ __launch_bounds__(256) void aff_kernel(const float* __restrict__ emb,
                                                  const float* __restrict__ Nrm,
                                                  float* __restrict__ out,
                                                  int skipInterior) {
  extern __shared__ float lds[];
  const int tid = threadIdx.x;
  const int w0 = blockIdx.x * 64;
  const int h0 = blockIdx.y * 16;
  const int d  = blockIdx.z;

  if (skipInterior) {
    const bool interior = (w0 == 64 || w0 == 128) && (h0 >= 32) && (h0 <= 224) && (d >= 1);
    if (interior) return;  // handled by aff_interior
  }

  if constexpr (USE_N) stage_norms<true>(Nrm, lds, d, h0, w0, tid);

  const int h = h0 + (tid >> 4);
  const int w = w0 + (tid & 15) * 4;
  const uint32_t sbase = (uint32_t)d * DSTRIDE + (uint32_t)h * HSTRIDE + (uint32_t)w;

  const float* cp[16];
  v2f a01[16], a23[16];
#pragma unroll
  for (int c = 0; c < 16; ++c) {
    cp[c] = emb + sbase + (uint32_t)c * CSTRIDE;
    const f4u t = *reinterpret_cast<const f4u*>(cp[c]);
    a01[c] = v2f{t.x, t.y};
    a23[c] = v2f{t.z, t.w};
  }

  v2f halfN01 = {0.f, 0.f}, halfN23 = {0.f, 0.f};
  if constexpr (USE_N) {
    const int sb = ((tid >> 4) + 27) * R0_STRIDE + ((tid & 15) * 4 + 27);
    halfN01 = v2f{0.5f * lds[sb + 0], 0.5f * lds[sb + 1]};
    halfN23 = v2f{0.5f * lds[sb + 2], 0.5f * lds[sb + 3]};
  }

#pragma unroll
  for (int k = 0; k < NOFF; ++k) {
    const int dd = c_DD[k], dh = c_DH[k], dwo = c_DW[k];
    const float coef = (USE_N ? 0.5f : 0.25f) / c_S2[k];
    const int delta = dd * (int)DSTRIDE + dh * (int)HSTRIDE + dwo;  // constexpr after unroll
    const int dn = d + dd, hn = h + dh, wn = w + dwo;
    const bool phv = ((unsigned)dn < 32u) && ((unsigned)hn < 256u);
    v4 aff;
    if (!phv) {
      aff.x = aff.y = aff.z = aff.w = 1.0f;
    } else if (wn >= 0 && wn <= 252) {
      v2f acc01, acc23;
      dot16p<USE_N>(cp, delta, a01, a23, acc01, acc23);
      if constexpr (USE_N) {
        const int bi = (dd == 0)
            ? ((hn - h0 + 27) * R0_STRIDE + (wn - w0 + 27))
            : (R0_SIZE + (hn - h0 + 1) * R1_STRIDE + (wn - w0 + 1));
        aff_fast_useN(lds, bi, halfN01, halfN23, acc01, acc23, coef, aff);
      } else {
        const v2f cf = {coef, coef};
        const v2f g01 = acc01 * cf, g23 = acc23 * cf;
        aff.x = aff_from_arg(g01.x); aff.y = aff_from_arg(g01.y);
        aff.z = aff_from_arg(g23.x); aff.w = aff_from_arg(g23.y);
      }
    } else {
      // -------- w-boundary: per-component scalar --------
#pragma unroll
      for (int j = 0; j < 4; ++j) {
        float av = 1.0f;
        const int wnj = wn + j;
        if ((unsigned)wnj < 256u) {
          float accj = 0.0f;
#pragma unroll
          for (int c = 0; c < 16; ++c) {
            const float bv = cp[c][delta + j];
            const float avx = (j < 2) ? ((j == 0) ? a01[c].x : a01[c].y)
                                      : ((j == 2) ? a23[c].x : a23[c].y);
            if constexpr (USE_N) {
              accj = fmaf(avx, bv, accj);
            } else {
              const float t = avx - bv;
              accj = fmaf(t, t, accj);
            }
          }
          if constexpr (USE_N) {
            const int bi = (dd == 0)
                ? ((hn - h0 + 27) * R0_STRIDE + (wnj - w0 + 27))
                : (R0_SIZE + (hn - h0 + 1) * R1_STRIDE + (wnj - w0 + 1));
            const float hNj = (j < 2) ? ((j == 0) ? halfN01.x : halfN01.y)
                                      : ((j == 2) ? halfN23.x : halfN23.y);
            const float sj = fmaf(0.5f, lds[bi], hNj);
            av = aff_from_arg((sj - accj) * coef);
          } else {
            av = aff_from_arg(accj * coef);
          }
        }
        setc(aff, j, av);
      }
    }
    store_nt4(out + (uint32_t)k * CSTRIDE + sbase, aff);
  }
}

extern "C" void kernel_launch(void* const* d_in, const int* in_sizes, int n_in,
                              void* d_out, int out_size, void* d_ws, size_t ws_size,
                              hipStream_t stream) {
  (void)in_sizes; (void)n_in; (void)out_size;
  const float* emb = (const float*)d_in[0];
  float* out = (float*)d_out;
  float* Nrm = (float*)d_ws;

  const dim3 gridAll(4, 16, 32);
  const size_t shmem = LDS_FLOATS * sizeof(float);
  const bool useN = (Nrm != nullptr) && (ws_size >= (size_t)CSTRIDE * sizeof(float));

  if (useN) {
    norm_kernel<<<CSTRIDE / 4 / 256, 256, 0, stream>>>(emb, Nrm);
    aff_interior<<<dim3(2, 13, 31), 256, shmem, stream>>>(emb, Nrm, out);
    aff_kernel<true><<<gridAll, 256, shmem, stream>>>(emb, Nrm, out, 1);
  } else {
    aff_kernel<false><<<gridAll, 256, 0, stream>>>(emb, nullptr, out, 0);
  }
}